// model_54451595379306
// MI455X (gfx1250) — compile-verified
//
#include <hip/hip_runtime.h>

// Problem dims (match reference)
#define T_   512
#define B_   128
#define DIN  512
#define DH   1024
#define M1   (T_ * B_)        // 65536 rows for the input projection GEMM
#define NWG  64               // persistent workgroups for the recurrence

typedef float v2f __attribute__((ext_vector_type(2)));
typedef float v8f __attribute__((ext_vector_type(8)));

// Feature-test the gfx1250 async global->LDS copy path (ASYNCcnt-tracked).
#if defined(__has_builtin)
#if __has_builtin(__builtin_amdgcn_global_load_async_to_lds_b32) && \
    __has_builtin(__builtin_amdgcn_s_wait_asynccnt)
#define USE_ASYNC_LDS 1
#endif
#endif

// f32 WMMA: D(16x16,f32) = A(16x4,f32) * B(4x16,f32) + C
// 8-arg form: (neg_a, A, neg_b, B, c_mod, C, reuse_a, reuse_b)
__device__ __forceinline__ v8f wmma4(v2f a, v2f b, v8f c) {
  return __builtin_amdgcn_wmma_f32_16x16x4_f32(false, a, false, b, (short)0, c,
                                               false, false);
}

// ---------------------------------------------------------------------------
// Phase 0: zero the per-step barrier counters (workspace is not re-poisoned
// between graph replays, so we must re-init every launch).
// ---------------------------------------------------------------------------
__global__ void zero_bar_kernel(int* __restrict__ bar, int n) {
  int i = blockIdx.x * blockDim.x + threadIdx.x;
  if (i < n) bar[i] = 0;
}

// ---------------------------------------------------------------------------
// Phase 1: xin[t,b,h] = sum_i x[t,b,i]*Wi_w[h,i] + Wi_b[h] + W1_b[h]
// Stored into d_out[(t*B+b)*DH + h] (pre-activation storage reused as state).
// Each wave computes a 16x64 strip (4 WMMA accumulators reuse each A load).
// ---------------------------------------------------------------------------
__global__ __launch_bounds__(256) void xin_gemm_kernel(
    const float* __restrict__ x, const float* __restrict__ Wi_w,
    const float* __restrict__ Wi_b, const float* __restrict__ W1_b,
    float* __restrict__ out) {
  const int wave = threadIdx.x >> 5;
  const int lane = threadIdx.x & 31;
  const int strip = blockIdx.x * 8 + wave;     // 65536 strips total
  const int nStrip = strip & 15;               // 16 strips of 64 cols
  const int mTile = strip >> 4;                // 4096 row tiles
  const int m0 = mTile * 16;
  const int n0 = nStrip * 64;

  const int ml = lane & 15;            // row (A) / col (B) within tile
  const int kb = (lane >> 4) << 1;     // K sub-offset: lanes 0-15 -> 0, 16-31 -> 2
  const int mhi = (lane >> 4) << 3;    // C/D row offset for upper half-wave

  v8f acc[4];
#pragma unroll
  for (int j = 0; j < 4; ++j) acc[j] = (v8f){0.f,0.f,0.f,0.f,0.f,0.f,0.f,0.f};

  const float* arow = x + (size_t)(m0 + ml) * DIN;
  for (int k0 = 0; k0 < DIN; k0 += 4) {
    v2f a = *(const v2f*)(arow + k0 + kb);
#pragma unroll
    for (int j = 0; j < 4; ++j) {
      // B[k][n] = Wi_w[n][k]; lane supplies (K=k0+kb, k0+kb+1) at N=n0+16j+ml
      v2f b = *(const v2f*)(Wi_w + (size_t)(n0 + 16 * j + ml) * DIN + k0 + kb);
      acc[j] = wmma4(a, b, acc[j]);
    }
  }

#pragma unroll
  for (int j = 0; j < 4; ++j) {
    const int n = n0 + 16 * j + ml;
    const float bias = Wi_b[n] + W1_b[n];
#pragma unroll
    for (int r = 0; r < 8; ++r) {
      const int m = m0 + r + mhi;
      out[(size_t)m * DH + n] = acc[j][r] + bias;
    }
  }
}

// ---------------------------------------------------------------------------
// Phase 2: persistent recurrence. 64 WGs, each owns 16 output columns and
// keeps its W1^T slice (1024x16 f32 = 64KB) resident in LDS for all 512
// steps. Per step: c = xin_t(own cols, already in d_out[t]) + h_{t-1} @ W1^T,
// written back into d_out[t]; device-wide atomic barrier between steps.
// ---------------------------------------------------------------------------
__global__ __launch_bounds__(256) void recurrence_kernel(
    const float* __restrict__ h0, const float* __restrict__ W1_w,
    float* __restrict__ out, int* __restrict__ bar, int nwg) {
  // LDS: W1^T slice, k-pair interleaved: ldsW[(k>>1)*32 + nn*2 + (k&1)]
  __shared__ float ldsW[DH * 16];  // 64 KB, resident for all 512 steps

  const int tid = threadIdx.x;
  const int n0 = blockIdx.x * 16;  // owned column slice

  // Cooperative load of W1^T slice (coalesced global reads along k),
  // via the ASYNCcnt-tracked global->LDS DMA path on gfx1250.
#if USE_ASYNC_LDS
  for (int idx = tid; idx < DH * 16; idx += 256) {
    const int nn = idx >> 10;        // 0..15
    const int k = idx & (DH - 1);    // 0..1023
    __builtin_amdgcn_global_load_async_to_lds_b32(
        (int*)&W1_w[(size_t)(n0 + nn) * DH + k],
        (int*)&ldsW[(k >> 1) * 32 + nn * 2 + (k & 1)],
        0, 0);
  }
  __builtin_amdgcn_s_wait_asynccnt(0);
  __syncthreads();
#else
  for (int idx = tid; idx < DH * 16; idx += 256) {
    const int nn = idx >> 10;        // 0..15
    const int k = idx & (DH - 1);    // 0..1023
    const float v = W1_w[(size_t)(n0 + nn) * DH + k];
    ldsW[(k >> 1) * 32 + nn * 2 + (k & 1)] = v;
  }
  __syncthreads();
#endif

  const int wave = tid >> 5;
  const int lane = tid & 31;
  const int m0 = wave * 16;            // 8 waves cover M = 128 rows
  const int ml = lane & 15;
  const int kb = (lane >> 4) << 1;
  const int mhi = (lane >> 4) << 3;
  const int n = n0 + ml;

  for (int t = 0; t < T_; ++t) {
    const float* prev = (t == 0) ? h0 : out + (size_t)(t - 1) * B_ * DH;
    float* cur = out + (size_t)t * B_ * DH;

    // c <- xin_t tile (this WG's own columns; only we overwrite them)
    v8f c;
#pragma unroll
    for (int r = 0; r < 8; ++r)
      c[r] = cur[(size_t)(m0 + r + mhi) * DH + n];

    // Prefetch next timestep's xin tile into cache while we compute.
    if (t + 1 < T_) {
      __builtin_prefetch(cur + (size_t)B_ * DH + (size_t)(m0 + ml) * DH + n0,
                         0, 3);
    }

    const float* arow = prev + (size_t)(m0 + ml) * DH;
#pragma unroll 16
    for (int k0 = 0; k0 < DH; k0 += 4) {
      v2f a = *(const v2f*)(arow + k0 + kb);
      v2f b = *(const v2f*)(&ldsW[((k0 + kb) >> 1) * 32 + ml * 2]);
      c = wmma4(a, b, c);
    }

#pragma unroll
    for (int r = 0; r < 8; ++r)
      cur[(size_t)(m0 + r + mhi) * DH + n] = c[r];

    // Publish h_t, then device-wide barrier on step t.
    __threadfence();
    __syncthreads();
    if (tid == 0) {
      __hip_atomic_fetch_add(&bar[t], 1, __ATOMIC_RELEASE,
                             __HIP_MEMORY_SCOPE_AGENT);
      while (__hip_atomic_load(&bar[t], __ATOMIC_ACQUIRE,
                               __HIP_MEMORY_SCOPE_AGENT) < nwg) {
        __builtin_amdgcn_s_sleep(1);
      }
    }
    __syncthreads();
  }

  // h_final (pre-sigmoid) for our columns.
  for (int idx = tid; idx < B_ * 16; idx += 256) {
    const int m = idx >> 4;
    const int nn = idx & 15;
    out[(size_t)T_ * B_ * DH + (size_t)m * DH + n0 + nn] =
        out[((size_t)(T_ - 1) * B_ + m) * DH + n0 + nn];
  }
  __syncthreads();

  // outs = sigmoid(h_t) in place, our columns only (we wrote them all).
  for (size_t idx = tid; idx < (size_t)T_ * B_ * 16; idx += 256) {
    const size_t t = idx / (B_ * 16);
    const int rem = (int)(idx % (B_ * 16));
    const int m = rem >> 4;
    const int nn = rem & 15;
    const size_t off = ((size_t)t * B_ + m) * DH + n0 + nn;
    const float v = out[off];
    out[off] = 1.0f / (1.0f + __expf(-v));
  }
}

// ---------------------------------------------------------------------------
extern "C" void kernel_launch(void* const* d_in, const int* in_sizes, int n_in,
                              void* d_out, int out_size, void* d_ws,
                              size_t ws_size, hipStream_t stream) {
  const float* x    = (const float*)d_in[0];  // [T,B,DIN]
  const float* h0   = (const float*)d_in[1];  // [B,DH]
  const float* Wi_w = (const float*)d_in[2];  // [DH,DIN]
  const float* Wi_b = (const float*)d_in[3];  // [DH]
  const float* W1_w = (const float*)d_in[4];  // [DH,DH]
  const float* W1_b = (const float*)d_in[5];  // [DH]
  float* out = (float*)d_out;                 // outs[T,B,DH] ++ h_final[B,DH]
  int* bar = (int*)d_ws;                      // T barrier counters

  zero_bar_kernel<<<(T_ + 255) / 256, 256, 0, stream>>>(bar, T_);

  // 65536 strips / 8 waves per block
  xin_gemm_kernel<<<(M1 / 16) * (DH / 64) / 8, 256, 0, stream>>>(
      x, Wi_w, Wi_b, W1_b, out);

  recurrence_kernel<<<NWG, 256, 0, stream>>>(h0, W1_w, out, bar, NWG);
}